// Mask_Oneshot_Layer_4002909520584
// MI455X (gfx1250) — compile-verified
//
#include <hip/hip_runtime.h>
#include <math.h>

// ---------------------------------------------------------------------------
// Problem constants
// ---------------------------------------------------------------------------
#define BATCH 64
#define NN    256            // H == W == 256
#define HW    (NN*NN)        // 65536
#define SPARSITY 0.125f
#define PMASK_SLOPE 5.0f

// ws layout (float indices)
#define WS_C    0                      // 256x256 cos twiddles
#define WS_S    65536                  // 256x256 (-sin) twiddles
#define WS_ACC  131072                 // 1 float accumulator
#define WS_TRE  131328                 // BATCH*HW intermediate (re)
#define WS_TIM  (131328 + BATCH*HW)    // BATCH*HW intermediate (im)

// d_out layout (float offsets), tuple return order:
// (uifft[B,2,H,W], complex_abs[B,1,H,W], mask[B,1,H,W], fft[B,2,H,W], u_k[B,2,H,W])
#define OUT_UIFFT 0
#define OUT_ABS   (2*BATCH*HW)
#define OUT_MASK  (OUT_ABS + BATCH*HW)
#define OUT_FFT   (OUT_MASK + BATCH*HW)
#define OUT_UK    (OUT_FFT + 2*BATCH*HW)

typedef float v2f __attribute__((ext_vector_type(2)));
typedef float v8f __attribute__((ext_vector_type(8)));

// Full-precision f32 WMMA: D(16x16) += A(16x4) * B(4x16)
// 8-arg form: (neg_a, A, neg_b, B, c_mod, C, reuse_a, reuse_b)
__device__ __forceinline__ v8f wmma4(v2f a, v2f b, v8f c) {
  return __builtin_amdgcn_wmma_f32_16x16x4_f32(false, a, false, b, (short)0, c,
                                               false, false);
}

// ---------------------------------------------------------------------------
// Kernel 0a: twiddle factors  C[j,k]=cos(2*pi*j*k/256), S[j,k]=-sin(...)
//            (i.e. exp(-i*theta) = C + i*S).  Also zero the mean accumulator.
// ---------------------------------------------------------------------------
__global__ __launch_bounds__(256) void k_twiddle(float* __restrict__ ws) {
  const int j = blockIdx.x;
  const int k = threadIdx.x;
  const int p = (j * k) & (NN - 1);            // phase mod 256 for precision
  const float ang = -6.28318530717958647692f * (float)p * (1.0f / (float)NN);
  float s, c;
  sincosf(ang, &s, &c);
  ws[WS_C + j * NN + k] = c;
  ws[WS_S + j * NN + k] = s;
  if (j == 0 && k == 0) ws[WS_ACC] = 0.0f;
}

// ---------------------------------------------------------------------------
// Kernel 0b: mean of sigmoid(5*weight) over H*W  (block reduce + atomicAdd)
// ---------------------------------------------------------------------------
__global__ __launch_bounds__(256) void k_mean(const float* __restrict__ weight,
                                              float* __restrict__ ws) {
  __shared__ float sdata[256];
  const int i = blockIdx.x * 256 + threadIdx.x;
  const float w = weight[i];
  sdata[threadIdx.x] = 1.0f / (1.0f + __expf(-PMASK_SLOPE * w));
  __syncthreads();
  for (int off = 128; off > 0; off >>= 1) {
    if (threadIdx.x < off) sdata[threadIdx.x] += sdata[threadIdx.x + off];
    __syncthreads();
  }
  if (threadIdx.x == 0) atomicAdd(&ws[WS_ACC], sdata[0]);
}

// ---------------------------------------------------------------------------
// Kernel 0c: rescale prob map + stochastic threshold -> binary mask [B,1,H,W]
// ---------------------------------------------------------------------------
__device__ __forceinline__ float urand_hash(unsigned int idx) {
  unsigned int s = idx * 747796405u + 2891336453u;
  unsigned int w = ((s >> ((s >> 28) + 4u)) ^ s) * 277803737u;
  w = (w >> 22) ^ w;
  return (float)(w >> 8) * (1.0f / 16777216.0f);   // [0,1)
}

__global__ __launch_bounds__(256) void k_mask(const float* __restrict__ weight,
                                              const float* __restrict__ ws,
                                              float* __restrict__ out_mask) {
  const unsigned int idx = blockIdx.x * 256 + threadIdx.x;  // over B*H*W
  const int hw = (int)(idx & (HW - 1));
  const float w = weight[hw];
  const float p1 = 1.0f / (1.0f + __expf(-PMASK_SLOPE * w));
  const float xbar = ws[WS_ACC] * (1.0f / (float)HW);
  const float r = SPARSITY / xbar;
  const float beta = (1.0f - SPARSITY) / (1.0f - xbar);
  const float p2 = (r < 1.0f) ? (p1 * r) : (1.0f - (1.0f - p1) * beta);
  const float u = urand_hash(idx * 2654435761u + 12345u);
  out_mask[idx] = (p2 > u) ? 1.0f : 0.0f;
}

// ---------------------------------------------------------------------------
// Tile mapping: one wave -> one 16x32 output tile (two 16x16 n-tiles sharing
// the A fragments).  K loop over 256 in steps of 4 with V_WMMA_F32_16X16X4_F32.
//   A frag (16x4):  lane m = lane&15, kk = 2*(lane>>4); a = {A[m][k+kk], A[m][k+kk+1]}
//   B frag (4x16):  lane n = lane&15; b = {B[k+kk][n],  B[k+kk+1][n]}
//   D (16x16):      elem v -> row = 16*tm + v + 8*(lane>>4), col = 16*tn + (lane&15)
// Waves per image per GEMM: 16 (tm) * 8 (tn pairs) = 128;  total = 64*128 = 8192.
// ---------------------------------------------------------------------------
struct TileIdx { int b, arow, ncol0, ncol1, rbase; };

__device__ __forceinline__ TileIdx tile_idx() {
  TileIdx t;
  const int wave = threadIdx.x >> 5;
  const int g = blockIdx.x * 8 + wave;   // global wave-tile id: BATCH*128 total
  t.b = g >> 7;
  const int tt = g & 127;
  const int tm = tt >> 3;                // 16 row-blocks
  const int tn2 = tt & 7;                // 8 column double-blocks (32 cols each)
  const int lane = threadIdx.x & 31;
  const int mlo = lane & 15;
  const int kk = (lane >> 4) << 1;
  t.arow = (tm * 16 + mlo) * NN + kk;    // A-fragment base (add k0)
  t.ncol0 = tn2 * 32 + mlo;
  t.ncol1 = t.ncol0 + 16;
  t.rbase = tm * 16 + 8 * (lane >> 4);   // D row base (add v)
  return t;
}

// ---------------------------------------------------------------------------
// Kernel 1: FFT stage 1.  T = F * X  (A = C/S twiddles, B = real input X)
// ---------------------------------------------------------------------------
__global__ __launch_bounds__(256) void k_fft1(const float* __restrict__ x,
                                              float* __restrict__ ws) {
  const TileIdx t = tile_idx();
  const float* C = ws + WS_C;
  const float* S = ws + WS_S;
  const float* X = x + t.b * HW;
  v8f accRe0 = {}, accIm0 = {}, accRe1 = {}, accIm1 = {};
#pragma unroll 4
  for (int k0 = 0; k0 < NN; k0 += 4) {
    v2f aC = *(const v2f*)(C + t.arow + k0);
    v2f aS = *(const v2f*)(S + t.arow + k0);
    const float* Xr0 = X + (k0 + ((t.arow) & 0)) ;  // (dummy avoid warn)
    (void)Xr0;
    const int kr = k0 + ((threadIdx.x & 31) >> 4) * 2;
    v2f bX0, bX1;
    bX0.x = X[kr * NN + t.ncol0];  bX0.y = X[(kr + 1) * NN + t.ncol0];
    bX1.x = X[kr * NN + t.ncol1];  bX1.y = X[(kr + 1) * NN + t.ncol1];
    accRe0 = wmma4(aC, bX0, accRe0);
    accIm0 = wmma4(aS, bX0, accIm0);
    accRe1 = wmma4(aC, bX1, accRe1);
    accIm1 = wmma4(aS, bX1, accIm1);
  }
  float* Tre = ws + WS_TRE + t.b * HW;
  float* Tim = ws + WS_TIM + t.b * HW;
#pragma unroll
  for (int v = 0; v < 8; ++v) {
    const int r = (t.rbase + v) * NN;
    Tre[r + t.ncol0] = accRe0[v];
    Tim[r + t.ncol0] = accIm0[v];
    Tre[r + t.ncol1] = accRe1[v];
    Tim[r + t.ncol1] = accIm1[v];
  }
}

// ---------------------------------------------------------------------------
// Kernel 2: FFT stage 2 + mask apply.
//   Yre = Tre*C - Tim*S ; Yim = Tre*S + Tim*C   (right-multiply by F)
//   writes fft[B,2,H,W] and u_k = fft * mask
// ---------------------------------------------------------------------------
__global__ __launch_bounds__(256) void k_fft2(const float* __restrict__ ws,
                                              float* __restrict__ out) {
  const TileIdx t = tile_idx();
  const float* C = ws + WS_C;
  const float* S = ws + WS_S;
  const float* Tre = ws + WS_TRE + t.b * HW;
  const float* Tim = ws + WS_TIM + t.b * HW;
  v8f accRe0 = {}, accIm0 = {}, accRe1 = {}, accIm1 = {};
#pragma unroll 2
  for (int k0 = 0; k0 < NN; k0 += 4) {
    v2f aRe = *(const v2f*)(Tre + t.arow + k0);
    v2f aIm = *(const v2f*)(Tim + t.arow + k0);
    v2f aImN; aImN.x = -aIm.x; aImN.y = -aIm.y;
    const int kr = k0 + ((threadIdx.x & 31) >> 4) * 2;
    v2f bC0, bS0, bC1, bS1;
    bC0.x = C[kr * NN + t.ncol0];  bC0.y = C[(kr + 1) * NN + t.ncol0];
    bS0.x = S[kr * NN + t.ncol0];  bS0.y = S[(kr + 1) * NN + t.ncol0];
    bC1.x = C[kr * NN + t.ncol1];  bC1.y = C[(kr + 1) * NN + t.ncol1];
    bS1.x = S[kr * NN + t.ncol1];  bS1.y = S[(kr + 1) * NN + t.ncol1];
    accRe0 = wmma4(aRe, bC0, accRe0);
    accRe0 = wmma4(aImN, bS0, accRe0);
    accIm0 = wmma4(aRe, bS0, accIm0);
    accIm0 = wmma4(aIm, bC0, accIm0);
    accRe1 = wmma4(aRe, bC1, accRe1);
    accRe1 = wmma4(aImN, bS1, accRe1);
    accIm1 = wmma4(aRe, bS1, accIm1);
    accIm1 = wmma4(aIm, bC1, accIm1);
  }
  const float* maskp = out + OUT_MASK + t.b * HW;
  float* fre = out + OUT_FFT + t.b * 2 * HW;
  float* fim = fre + HW;
  float* ure = out + OUT_UK + t.b * 2 * HW;
  float* uim = ure + HW;
#pragma unroll
  for (int v = 0; v < 8; ++v) {
    const int r = (t.rbase + v) * NN;
    {
      const int off = r + t.ncol0;
      const float m = maskp[off];
      const float re = accRe0[v], im = accIm0[v];
      fre[off] = re;  fim[off] = im;  ure[off] = re * m;  uim[off] = im * m;
    }
    {
      const int off = r + t.ncol1;
      const float m = maskp[off];
      const float re = accRe1[v], im = accIm1[v];
      fre[off] = re;  fim[off] = im;  ure[off] = re * m;  uim[off] = im * m;
    }
  }
}

// ---------------------------------------------------------------------------
// Kernel 3: iFFT stage 1.  Z = conj(F) * U  (A = C/S, B = u_k re/im)
//   Zre = C*Ure + S*Uim ; Zim = C*Uim - S*Ure
// ---------------------------------------------------------------------------
__global__ __launch_bounds__(256) void k_ifft1(const float* __restrict__ out,
                                               float* __restrict__ ws) {
  const TileIdx t = tile_idx();
  const float* C = ws + WS_C;
  const float* S = ws + WS_S;
  const float* Ure = out + OUT_UK + t.b * 2 * HW;
  const float* Uim = Ure + HW;
  v8f accRe0 = {}, accIm0 = {}, accRe1 = {}, accIm1 = {};
#pragma unroll 2
  for (int k0 = 0; k0 < NN; k0 += 4) {
    v2f aC = *(const v2f*)(C + t.arow + k0);
    v2f aS = *(const v2f*)(S + t.arow + k0);
    const int kr = k0 + ((threadIdx.x & 31) >> 4) * 2;
    v2f bRe0, bIm0, bRe1, bIm1;
    bRe0.x = Ure[kr * NN + t.ncol0];  bRe0.y = Ure[(kr + 1) * NN + t.ncol0];
    bIm0.x = Uim[kr * NN + t.ncol0];  bIm0.y = Uim[(kr + 1) * NN + t.ncol0];
    bRe1.x = Ure[kr * NN + t.ncol1];  bRe1.y = Ure[(kr + 1) * NN + t.ncol1];
    bIm1.x = Uim[kr * NN + t.ncol1];  bIm1.y = Uim[(kr + 1) * NN + t.ncol1];
    v2f bRe0N; bRe0N.x = -bRe0.x; bRe0N.y = -bRe0.y;
    v2f bRe1N; bRe1N.x = -bRe1.x; bRe1N.y = -bRe1.y;
    accRe0 = wmma4(aC, bRe0, accRe0);
    accRe0 = wmma4(aS, bIm0, accRe0);
    accIm0 = wmma4(aC, bIm0, accIm0);
    accIm0 = wmma4(aS, bRe0N, accIm0);
    accRe1 = wmma4(aC, bRe1, accRe1);
    accRe1 = wmma4(aS, bIm1, accRe1);
    accIm1 = wmma4(aC, bIm1, accIm1);
    accIm1 = wmma4(aS, bRe1N, accIm1);
  }
  float* Zre = ws + WS_TRE + t.b * HW;
  float* Zim = ws + WS_TIM + t.b * HW;
#pragma unroll
  for (int v = 0; v < 8; ++v) {
    const int r = (t.rbase + v) * NN;
    Zre[r + t.ncol0] = accRe0[v];
    Zim[r + t.ncol0] = accIm0[v];
    Zre[r + t.ncol1] = accRe1[v];
    Zim[r + t.ncol1] = accIm1[v];
  }
}

// ---------------------------------------------------------------------------
// Kernel 4: iFFT stage 2 + 1/N^2 scale + complex magnitude.
//   Yre = Zre*C + Zim*S ; Yim = Zim*C - Zre*S   (right-multiply by conj(F))
// ---------------------------------------------------------------------------
__global__ __launch_bounds__(256) void k_ifft2(const float* __restrict__ ws,
                                               float* __restrict__ out) {
  const TileIdx t = tile_idx();
  const float* C = ws + WS_C;
  const float* S = ws + WS_S;
  const float* Zre = ws + WS_TRE + t.b * HW;
  const float* Zim = ws + WS_TIM + t.b * HW;
  v8f accRe0 = {}, accIm0 = {}, accRe1 = {}, accIm1 = {};
#pragma unroll 2
  for (int k0 = 0; k0 < NN; k0 += 4) {
    v2f aRe = *(const v2f*)(Zre + t.arow + k0);
    v2f aIm = *(const v2f*)(Zim + t.arow + k0);
    v2f aReN; aReN.x = -aRe.x; aReN.y = -aRe.y;
    const int kr = k0 + ((threadIdx.x & 31) >> 4) * 2;
    v2f bC0, bS0, bC1, bS1;
    bC0.x = C[kr * NN + t.ncol0];  bC0.y = C[(kr + 1) * NN + t.ncol0];
    bS0.x = S[kr * NN + t.ncol0];  bS0.y = S[(kr + 1) * NN + t.ncol0];
    bC1.x = C[kr * NN + t.ncol1];  bC1.y = C[(kr + 1) * NN + t.ncol1];
    bS1.x = S[kr * NN + t.ncol1];  bS1.y = S[(kr + 1) * NN + t.ncol1];
    accRe0 = wmma4(aRe, bC0, accRe0);
    accRe0 = wmma4(aIm, bS0, accRe0);
    accIm0 = wmma4(aIm, bC0, accIm0);
    accIm0 = wmma4(aReN, bS0, accIm0);
    accRe1 = wmma4(aRe, bC1, accRe1);
    accRe1 = wmma4(aIm, bS1, accRe1);
    accIm1 = wmma4(aIm, bC1, accIm1);
    accIm1 = wmma4(aReN, bS1, accIm1);
  }
  float* yre = out + OUT_UIFFT + t.b * 2 * HW;
  float* yim = yre + HW;
  float* ab = out + OUT_ABS + t.b * HW;
  const float inv = 1.0f / (float)HW;
#pragma unroll
  for (int v = 0; v < 8; ++v) {
    const int r = (t.rbase + v) * NN;
    {
      const int off = r + t.ncol0;
      const float re = accRe0[v] * inv;
      const float im = accIm0[v] * inv;
      yre[off] = re;  yim[off] = im;  ab[off] = sqrtf(re * re + im * im);
    }
    {
      const int off = r + t.ncol1;
      const float re = accRe1[v] * inv;
      const float im = accIm1[v] * inv;
      yre[off] = re;  yim[off] = im;  ab[off] = sqrtf(re * re + im * im);
    }
  }
}

// ---------------------------------------------------------------------------
// Launch
// ---------------------------------------------------------------------------
extern "C" void kernel_launch(void* const* d_in, const int* in_sizes, int n_in,
                              void* d_out, int out_size, void* d_ws, size_t ws_size,
                              hipStream_t stream) {
  const float* x = (const float*)d_in[0];        // [B,1,H,W] f32
  const float* weight = (const float*)d_in[1];   // [1,1,H,W] f32
  float* out = (float*)d_out;
  float* ws = (float*)d_ws;

  (void)in_sizes; (void)n_in; (void)out_size; (void)ws_size;

  const int gemm_blocks = (BATCH * 128) / 8;     // 8 waves/block, 16x32 tile/wave

  k_twiddle<<<NN, 256, 0, stream>>>(ws);
  k_mean<<<HW / 256, 256, 0, stream>>>(weight, ws);
  k_mask<<<(BATCH * HW) / 256, 256, 0, stream>>>(weight, ws, out + OUT_MASK);
  k_fft1<<<gemm_blocks, 256, 0, stream>>>(x, ws);
  k_fft2<<<gemm_blocks, 256, 0, stream>>>(ws, out);
  k_ifft1<<<gemm_blocks, 256, 0, stream>>>(out, ws);
  k_ifft2<<<gemm_blocks, 256, 0, stream>>>(ws, out);
}